// GraphSAGEModel_43344809951941
// MI455X (gfx1250) — compile-verified
//
#include <hip/hip_runtime.h>

// GraphSAGE (2-layer) for MI455X / gfx1250, wave32.
// out = (relu(x@Ws0 + mean0@Wn0 + b0)) @ Ws1 + mean1@Wn1 + b1
// fp32 WMMA (V_WMMA_F32_16X16X4_F32) keeps full fp32 numerics vs the reference.
// mean is formed directly by the scatter: agg[dst] += h[src] * deg_inv[dst].

typedef __attribute__((ext_vector_type(2))) float v2f;
typedef __attribute__((ext_vector_type(8))) float v8f;

#define D_DIM 1024
#define BM 128
#define BN 128
#define BK 32
#define KBLKS (D_DIM / BK)   // 32 K-blocks per phase
#define LDA 36               // 32 + 4 pad: float4-aligned, spreads LDS banks
#define LDB 36
#define NTHREADS 256

__global__ void fill_zero_kernel(float* __restrict__ p, size_t n) {
  size_t i = (size_t)blockIdx.x * blockDim.x + threadIdx.x;
  size_t stride = (size_t)gridDim.x * blockDim.x;
  for (; i < n; i += stride) p[i] = 0.0f;
}

__global__ void degree_kernel(const int* __restrict__ dst, float* __restrict__ deg, int E) {
  int e = blockIdx.x * blockDim.x + threadIdx.x;
  if (e < E) atomicAdd(deg + dst[e], 1.0f);
}

__global__ void recip_kernel(float* __restrict__ deg, int N) {
  int i = blockIdx.x * blockDim.x + threadIdx.x;
  if (i < N) deg[i] = 1.0f / fmaxf(deg[i], 1.0f);
}

// One block per edge: gather h[src] (float4), scatter-add h[src]*deg_inv[dst]
// into mean[dst]. mean (102 MB) fits the 192 MB L2 -> atomics stay L2-resident.
__global__ __launch_bounds__(NTHREADS)
void scatter_mean_kernel(const float* __restrict__ h,
                         const int* __restrict__ src,
                         const int* __restrict__ dst,
                         const float* __restrict__ deginv,
                         float* __restrict__ mean) {
  int e = blockIdx.x;
  int s = src[e];
  int d = dst[e];
  float w = deginv[d];  // uniform across block -> broadcast load
  const float4 v = ((const float4*)(h + (size_t)s * D_DIM))[threadIdx.x];
  float* ap = mean + (size_t)d * D_DIM + (size_t)threadIdx.x * 4;
  atomicAdd(ap + 0, v.x * w);
  atomicAdd(ap + 1, v.y * w);
  atomicAdd(ap + 2, v.z * w);
  atomicAdd(ap + 3, v.w * w);
}

// Fused SAGE layer GEMM:
//   Out = Aself @ Wself + Amean @ Wneigh + bias   (+ optional ReLU)
// One K = 2*1024 accumulation over two (A,B) source phases.
// 128x128 tile per 256-thread block; 8 waves (2x4); each wave owns 4(M)x2(N)
// WMMA 16x16 f32 tiles. Register-prefetch software pipeline: tile t+1's global
// loads are issued before tile t's 64 WMMAs, so s_wait_loadcnt lands a full
// compute block after issue.
//
// f32 WMMA fragment layout (ISA 7.12.2): A 16x4 -> 2 VGPRs, lanes 0-15 hold
// M=lane with K=0,1; lanes 16-31 hold K=2,3. B 4x16 mirrored. Storing B
// K-contiguous (transposed) in LDS makes both fragments one aligned float2.
__global__ __launch_bounds__(NTHREADS)
void sage_gemm_kernel(const float* __restrict__ Aself,   // [N, D]
                      const float* __restrict__ Amean,   // [N, D]
                      const float* __restrict__ Wself,   // [D, D]
                      const float* __restrict__ Wneigh,  // [D, D]
                      const float* __restrict__ bias,    // [D]
                      float* __restrict__ Out,           // [N, D]
                      int Nrows, int do_relu) {
  __shared__ float lds_a[BM * LDA];
  __shared__ float lds_b[BN * LDB];   // transposed: [col][k]

  const int tid    = threadIdx.x;
  const int lane   = tid & 31;
  const int wave   = tid >> 5;   // 0..7
  const int waveM  = wave >> 2;  // 0..1  -> 64 rows each
  const int waveN  = wave & 3;   // 0..3  -> 32 cols each
  const int half   = lane >> 4;  // 0/1
  const int lane16 = lane & 15;

  const int m0 = blockIdx.y * BM;
  const int n0 = blockIdx.x * BN;

  // Per-thread staging coordinates (constant across iterations).
  int a_row[4], a_c4[4], b_k[4], b_c4[4];
#pragma unroll
  for (int i = 0; i < 4; ++i) {
    int linear = tid + i * NTHREADS;   // 0..1023
    a_row[i] = linear >> 3;            // 0..127
    a_c4[i]  = (linear & 7) << 2;      // 0,4,...,28
    b_k[i]   = linear >> 5;            // 0..31
    b_c4[i]  = (linear & 31) << 2;     // 0..124
  }

  float4 ra[4], rb[4];
  auto fetch = [&](int it) {
    const float* __restrict__ Aptr = (it < KBLKS) ? Aself : Amean;
    const float* __restrict__ Bptr = (it < KBLKS) ? Wself : Wneigh;
    const int kb = (it & (KBLKS - 1)) * BK;
#pragma unroll
    for (int i = 0; i < 4; ++i) {
      int grow = m0 + a_row[i];
      ra[i] = make_float4(0.f, 0.f, 0.f, 0.f);
      if (grow < Nrows)
        ra[i] = *(const float4*)(Aptr + (size_t)grow * D_DIM + kb + a_c4[i]);
    }
#pragma unroll
    for (int i = 0; i < 4; ++i)
      rb[i] = *(const float4*)(Bptr + (size_t)(kb + b_k[i]) * D_DIM + n0 + b_c4[i]);
  };

  v8f acc[4][2] = {};

  fetch(0);
  for (int it = 0; it < 2 * KBLKS; ++it) {
    if (it) __syncthreads();  // previous compute done before LDS overwrite
    // Commit prefetched registers to LDS.
#pragma unroll
    for (int i = 0; i < 4; ++i)
      *(float4*)(&lds_a[a_row[i] * LDA + a_c4[i]]) = ra[i];
#pragma unroll
    for (int i = 0; i < 4; ++i) {
      lds_b[(b_c4[i] + 0) * LDB + b_k[i]] = rb[i].x;
      lds_b[(b_c4[i] + 1) * LDB + b_k[i]] = rb[i].y;
      lds_b[(b_c4[i] + 2) * LDB + b_k[i]] = rb[i].z;
      lds_b[(b_c4[i] + 3) * LDB + b_k[i]] = rb[i].w;
    }
    __syncthreads();

    // Issue next tile's global loads; latency hides behind 64 WMMAs.
    if (it + 1 < 2 * KBLKS) fetch(it + 1);

    // 8 k-steps of K=4, 8 WMMAs each.
#pragma unroll
    for (int k4 = 0; k4 < BK / 4; ++k4) {
      const int koff = (k4 << 2) + (half << 1);  // even -> 8B aligned
      v2f afrag[4], bfrag[2];
#pragma unroll
      for (int mi = 0; mi < 4; ++mi)
        afrag[mi] = *(const v2f*)(&lds_a[(waveM * 64 + mi * 16 + lane16) * LDA + koff]);
#pragma unroll
      for (int ni = 0; ni < 2; ++ni)
        bfrag[ni] = *(const v2f*)(&lds_b[(waveN * 32 + ni * 16 + lane16) * LDB + koff]);
#pragma unroll
      for (int mi = 0; mi < 4; ++mi)
#pragma unroll
        for (int ni = 0; ni < 2; ++ni)
          acc[mi][ni] = __builtin_amdgcn_wmma_f32_16x16x4_f32(
              false, afrag[mi], false, bfrag[ni],
              (short)0, acc[mi][ni], false, false);
    }
  }

  // Epilogue: bias (+ReLU), masked store. Lanes 0-15 cover consecutive cols
  // per VGPR -> 64B-coalesced half-wave stores.
#pragma unroll
  for (int ni = 0; ni < 2; ++ni) {
    int col = n0 + waveN * 32 + ni * 16 + lane16;
    float bv = bias[col];
#pragma unroll
    for (int mi = 0; mi < 4; ++mi) {
#pragma unroll
      for (int v = 0; v < 8; ++v) {
        int row = m0 + waveM * 64 + mi * 16 + half * 8 + v;
        if (row < Nrows) {
          float r = acc[mi][ni][v] + bv;
          if (do_relu) r = fmaxf(r, 0.0f);
          Out[(size_t)row * D_DIM + col] = r;
        }
      }
    }
  }
}

extern "C" void kernel_launch(void* const* d_in, const int* in_sizes, int n_in,
                              void* d_out, int out_size, void* d_ws, size_t ws_size,
                              hipStream_t stream) {
  const float* x    = (const float*)d_in[0];
  const int*   esrc = (const int*)d_in[1];
  const int*   edst = (const int*)d_in[2];
  const float* Ws0  = (const float*)d_in[3];
  const float* Wn0  = (const float*)d_in[4];
  const float* b0   = (const float*)d_in[5];
  const float* Ws1  = (const float*)d_in[6];
  const float* Wn1  = (const float*)d_in[7];
  const float* b1   = (const float*)d_in[8];
  float* out = (float*)d_out;

  const int N = in_sizes[0] / D_DIM;  // 25000
  const int E = in_sizes[1];          // 400000

  // Workspace layout: mean [N*D] | h [N*D] | deg/deg_inv [N]
  float* mean   = (float*)d_ws;
  float* h      = mean + (size_t)N * D_DIM;
  float* deginv = h + (size_t)N * D_DIM;

  // degrees -> deg_inv (computed once, shared by both layers, as in reference)
  fill_zero_kernel<<<256, 256, 0, stream>>>(deginv, (size_t)N);
  degree_kernel<<<(E + 255) / 256, 256, 0, stream>>>(edst, deginv, E);
  recip_kernel<<<(N + 255) / 256, 256, 0, stream>>>(deginv, N);

  dim3 ggrid(D_DIM / BN, (N + BM - 1) / BM);  // 8 x 196

  // layer 0: h = relu(x@Ws0 + mean(x)@Wn0 + b0)
  fill_zero_kernel<<<2048, 256, 0, stream>>>(mean, (size_t)N * D_DIM);
  scatter_mean_kernel<<<E, NTHREADS, 0, stream>>>(x, esrc, edst, deginv, mean);
  sage_gemm_kernel<<<ggrid, NTHREADS, 0, stream>>>(x, mean, Ws0, Wn0, b0, h, N, 1);

  // layer 1: out = h@Ws1 + mean(h)@Wn1 + b1
  fill_zero_kernel<<<2048, 256, 0, stream>>>(mean, (size_t)N * D_DIM);
  scatter_mean_kernel<<<E, NTHREADS, 0, stream>>>(h, esrc, edst, deginv, mean);
  sage_gemm_kernel<<<ggrid, NTHREADS, 0, stream>>>(h, mean, Ws1, Wn1, b1, out, N, 0);
}